// SpatioTemporal_map_80951543595347
// MI455X (gfx1250) — compile-verified
//
#include <hip/hip_runtime.h>

#define B_  4
#define T_  5
#define C_  64
#define H_  48
#define W_  48
#define F_  10
#define HW_ (H_ * W_)       // 2304
#define N_  (T_ * HW_)      // 11520

typedef __attribute__((ext_vector_type(2))) float v2f;
typedef __attribute__((ext_vector_type(8))) float v8f;

// ---------------------------------------------------------------------------
// Kernel 1: fused conv1+avgpool -> s[b,n], and g3 projection -> g3[b,n,f]
// s[b,n] = sum_c x[b,c,n] * mean_f(conv1_w[f,c]) + mean_f(conv1_b)
// g3[b,n,f] = sum_c x[b,c,n] * g3_w[f,c] + g3_b[f]
// ---------------------------------------------------------------------------
__global__ void __launch_bounds__(256)
k1_pre(const float* __restrict__ x,        // [B,T,C,H,W]
       const float* __restrict__ conv1_w,  // [F,C]
       const float* __restrict__ conv1_b,  // [F]
       const float* __restrict__ g3_w,     // [F,C]
       const float* __restrict__ g3_b,     // [F]
       float* __restrict__ s,              // [B,N]
       float* __restrict__ g3)             // [B,N,F]
{
    __shared__ float ls_wbar[C_];
    __shared__ float ls_g3w[F_ * C_];
    __shared__ float ls_g3b[F_];
    __shared__ float ls_bbar;

    const int tid = threadIdx.x;
    if (tid < C_) {
        float acc = 0.f;
        #pragma unroll
        for (int f = 0; f < F_; ++f) acc += conv1_w[f * C_ + tid];
        ls_wbar[tid] = acc * (1.0f / F_);
    }
    for (int i = tid; i < F_ * C_; i += blockDim.x) ls_g3w[i] = g3_w[i];
    if (tid < F_) ls_g3b[tid] = g3_b[tid];
    if (tid == 0) {
        float acc = 0.f;
        #pragma unroll
        for (int f = 0; f < F_; ++f) acc += conv1_b[f];
        ls_bbar = acc * (1.0f / F_);
    }
    __syncthreads();

    const int idx = blockIdx.x * blockDim.x + tid;   // flat (b, n)
    if (idx >= B_ * N_) return;
    const int b  = idx / N_;
    const int n  = idx - b * N_;
    const int t  = n / HW_;
    const int hw = n - t * HW_;

    const float* xp = x + ((size_t)b * T_ + t) * C_ * HW_ + hw;

    float accs = ls_bbar;
    float accg[F_];
    #pragma unroll
    for (int f = 0; f < F_; ++f) accg[f] = ls_g3b[f];

    for (int c = 0; c < C_; ++c) {
        const float v = xp[(size_t)c * HW_];
        accs += v * ls_wbar[c];
        #pragma unroll
        for (int f = 0; f < F_; ++f) accg[f] += v * ls_g3w[f * C_ + c];
    }

    s[idx] = accs;
    float* gp = g3 + (size_t)idx * F_;
    #pragma unroll
    for (int f = 0; f < F_; ++f) gp[f] = accg[f];
}

// ---------------------------------------------------------------------------
// Kernel 2: z[0:4, :] = s @ ffnn_w^T + bias via V_WMMA_F32_16X16X4_F32.
// One workgroup = one 16-column tile of z. 8 waves split K = 11520 into
// chunks of 1440 (360 WMMA steps each); partial 16x16 accumulators are
// reduced through LDS. M is padded 4 -> 16.
//
// Branch-free padding: lanes with m>=4 load row (m&3) (always valid, L0-hot)
// and multiply by a 0.0 mask -> no EXEC manipulation in the hot loop, EXEC
// stays all-1s as WMMA requires.
//
// W (531 MB) is streamed exactly once -> non-temporal loads so the stream
// does not evict s (184 KB, re-read by all 720 workgroups) from L2.
//
// A (16x4) lane layout: lanes 0-15 hold M=lane, K={0,1}; lanes 16-31 K={2,3}.
// B (4x16) mirrors it:  lanes 0-15 hold N=lane, K={0,1}; lanes 16-31 K={2,3}.
// C/D (16x16): VGPR r, lanes 0-15 -> M=r; lanes 16-31 -> M=r+8.
// ---------------------------------------------------------------------------
__global__ void __launch_bounds__(256)
k2_gemm(const float* __restrict__ s,     // [B,N] rows 0..3 valid
        const float* __restrict__ Wm,    // ffnn_w [N,N] row-major
        const float* __restrict__ bias,  // [N]
        float* __restrict__ z)           // [B,N]
{
    __shared__ float red[8][16][16];

    const int lane = threadIdx.x & 31;
    const int wave = threadIdx.x >> 5;
    const int i0   = blockIdx.x * 16;

    const int   m     = lane & 15;              // M row (A) / N col (B)
    const int   kb    = (lane >> 4) << 1;       // 0 or 2
    const float amask = (m < B_) ? 1.0f : 0.0f; // zero-pad rows 4..15

    const int KPW  = N_ / 8;                    // 1440 K per wave
    int       k    = wave * KPW;
    const int kend = k + KPW;

    const float* arow = s  + (size_t)(m & 3) * N_ + kb;   // wrapped, masked
    const float* brow = Wm + (size_t)(i0 + m) * N_ + kb;  // W row i0+(lane&15)

    v8f c = {};
    for (; k < kend; k += 32) {                 // 8 WMMA steps per group
        __builtin_prefetch(brow + k + 512, 0, 1);   // ~2KB ahead on W stream
        #pragma unroll
        for (int kk = 0; kk < 32; kk += 4) {
            v2f a = *(const v2f*)(arow + k + kk);
            a.x *= amask;
            a.y *= amask;
            const v2f bb =
                __builtin_nontemporal_load((const v2f*)(brow + k + kk));
            c = __builtin_amdgcn_wmma_f32_16x16x4_f32(
                    /*neg_a=*/false, a, /*neg_b=*/false, bb,
                    /*c_mod=*/(short)0, c, /*reuse_a=*/false, /*reuse_b=*/false);
        }
    }

    #pragma unroll
    for (int r = 0; r < 8; ++r) {
        const int mm = r + ((lane >> 4) << 3);   // M = r or r+8
        red[wave][mm][m] = c[r];
    }
    __syncthreads();

    const int tm = threadIdx.x >> 4;   // 0..15
    const int tn = threadIdx.x & 15;   // 0..15
    if (tm < B_) {
        float acc = 0.f;
        #pragma unroll
        for (int wv = 0; wv < 8; ++wv) acc += red[wv][tm][tn];
        z[(size_t)tm * N_ + i0 + tn] = acc + bias[i0 + tn];
    }
}

// ---------------------------------------------------------------------------
// Kernel 3: per-batch softmax statistics over N (max, sum of exp)
// ---------------------------------------------------------------------------
__global__ void __launch_bounds__(256)
k3_stats(const float* __restrict__ z, float* __restrict__ stats)
{
    __shared__ float sm[256];
    const int b   = blockIdx.x;
    const int tid = threadIdx.x;
    const float* zr = z + (size_t)b * N_;

    float mx = -3.402823e38f;
    for (int i = tid; i < N_; i += 256) mx = fmaxf(mx, zr[i]);
    sm[tid] = mx;
    __syncthreads();
    for (int off = 128; off > 0; off >>= 1) {
        if (tid < off) sm[tid] = fmaxf(sm[tid], sm[tid + off]);
        __syncthreads();
    }
    const float gmax = sm[0];
    __syncthreads();

    float se = 0.f;
    for (int i = tid; i < N_; i += 256) se += __expf(zr[i] - gmax);
    sm[tid] = se;
    __syncthreads();
    for (int off = 128; off > 0; off >>= 1) {
        if (tid < off) sm[tid] += sm[tid + off];
        __syncthreads();
    }
    if (tid == 0) {
        stats[b * 2 + 0] = gmax;
        stats[b * 2 + 1] = sm[0];
    }
}

// ---------------------------------------------------------------------------
// Kernel 4: gate = a0*relu(z) + a1*sigmoid(z) + a2*softmax(z);
// out[b,t,f,h,w] = gate[b,n] * g3[b,n,f]
// ---------------------------------------------------------------------------
__global__ void __launch_bounds__(256)
k4_out(const float* __restrict__ z,      // [B,N]
       const float* __restrict__ g3,     // [B,N,F]
       const float* __restrict__ stats,  // [B][max,sumexp]
       const float* __restrict__ aw,     // [3]
       float* __restrict__ out)          // [B,T,F,H,W]
{
    const int idx = blockIdx.x * blockDim.x + threadIdx.x;
    if (idx >= B_ * N_) return;
    const int b = idx / N_;
    const int n = idx - b * N_;

    const float zv = z[idx];
    const float a0 = aw[0], a1 = aw[1], a2 = aw[2];
    const float mx = stats[b * 2 + 0];
    const float se = stats[b * 2 + 1];

    const float relu = zv > 0.f ? zv : 0.f;
    const float sig  = 1.f / (1.f + __expf(-zv));
    const float sf   = __expf(zv - mx) / se;
    const float gate = a0 * relu + a1 * sig + a2 * sf;

    const int t  = n / HW_;
    const int hw = n - t * HW_;
    const float* gp = g3 + (size_t)idx * F_;
    float* op = out + ((size_t)b * T_ + t) * F_ * HW_ + hw;
    #pragma unroll
    for (int f = 0; f < F_; ++f) op[(size_t)f * HW_] = gate * gp[f];
}

// ---------------------------------------------------------------------------
extern "C" void kernel_launch(void* const* d_in, const int* in_sizes, int n_in,
                              void* d_out, int out_size, void* d_ws, size_t ws_size,
                              hipStream_t stream)
{
    const float* input   = (const float*)d_in[0];
    // d_in[1] = input1 : unused by the reference
    const float* aw      = (const float*)d_in[2];
    const float* conv1_w = (const float*)d_in[3];
    const float* conv1_b = (const float*)d_in[4];
    const float* g3_w    = (const float*)d_in[5];
    const float* g3_b    = (const float*)d_in[6];
    const float* ffnn_w  = (const float*)d_in[7];
    const float* ffnn_b  = (const float*)d_in[8];
    float* out = (float*)d_out;

    // Workspace layout (floats): s[4*11520] | z[4*11520] | stats[8] | g3[4*11520*10]
    float* s     = (float*)d_ws;
    float* zbuf  = s + (size_t)B_ * N_;
    float* stats = zbuf + (size_t)B_ * N_;
    float* g3    = stats + 8;

    const int blocksBN = (B_ * N_ + 255) / 256;   // 180

    k1_pre <<<blocksBN, 256, 0, stream>>>(input, conv1_w, conv1_b, g3_w, g3_b, s, g3);
    k2_gemm<<<N_ / 16, 256, 0, stream>>>(s, ffnn_w, ffnn_b, zbuf);
    k3_stats<<<B_, 256, 0, stream>>>(zbuf, stats);
    k4_out <<<blocksBN, 256, 0, stream>>>(zbuf, g3, stats, aw, out);
}